// WasserIndexGen_28132035788905
// MI455X (gfx1250) — compile-verified
//
#include <hip/hip_runtime.h>
#include <hip/hip_bf16.h>
#include <math.h>

// WasserIndexGen: Sinkhorn Wasserstein barycenter, N=4096 support, 64 topics.
// f32 WMMA (V_WMMA_F32_16X16X4_F32) GEMMs; C and C^T live in L2 (128MB < 192MB).
// Async global->LDS double-buffered staging (ASYNCcnt), K split 4-ways for
// occupancy. The GEMM B operand (u / t, 1MB) is kept k-pair-packed in global
// memory by its producer kernels so the WMMA B operand is a single aligned
// ds_load_b64 (no v_mov repacking in the hot loop).

typedef float v2f __attribute__((ext_vector_type(2)));
typedef float v8f __attribute__((ext_vector_type(8)));
typedef int   v4i __attribute__((ext_vector_type(4)));

#define NN     4096
#define NBB    64
#define ITERS  30
#define KSPLIT 4
#define CHUNK  64
#define CHUNKS ((NN / KSPLIT) / CHUNK)   // 16

// Pair-packed B layout: Bp[(k>>1)*128 + 2n + (k&1)]  (pair-row = 128 floats)
__device__ __forceinline__ size_t packIdx(int k, int n) {
  return (size_t)(k >> 1) * 128 + 2 * n + (k & 1);
}

// ---------------------------------------------------------------------------
// Async global->LDS copy (16B granules) + ASYNCcnt wait, with sync fallback.
// Builtin signature (from hipcc diagnostic): (v4i AS1*, v4i AS3*, Iint, Iint).
// ---------------------------------------------------------------------------
#if defined(__has_builtin)
#if __has_builtin(__builtin_amdgcn_global_load_async_to_lds_b128)
#define HAVE_ASYNC_LDS 1
#endif
#endif
#ifndef HAVE_ASYNC_LDS
#define HAVE_ASYNC_LDS 0
#endif

typedef __attribute__((address_space(1))) v4i* gptr4;
typedef __attribute__((address_space(3))) v4i* lptr4;

__device__ __forceinline__ void cp16(const float* __restrict__ g, float* l) {
#if HAVE_ASYNC_LDS
  __builtin_amdgcn_global_load_async_to_lds_b128((gptr4)g, (lptr4)l, 0, 0);
#else
  *(float4*)l = *(const float4*)g;
#endif
}

template <int N>
__device__ __forceinline__ void wait_async_le() {
#if HAVE_ASYNC_LDS
#if __has_builtin(__builtin_amdgcn_s_wait_asynccnt)
  __builtin_amdgcn_s_wait_asynccnt(N);
#else
  asm volatile("s_wait_asynccnt %0" ::"n"(N));
#endif
#endif
}

// ---------------------------------------------------------------------------
// C = exp(-M/reg) and CT = C^T, 64x64 tiles, LDS transpose (stride 65 pad).
// ---------------------------------------------------------------------------
__global__ __launch_bounds__(256) void build_c_ct(
    const float* __restrict__ M, const float* __restrict__ reg,
    float* __restrict__ C, float* __restrict__ CT)
{
  __shared__ float tile[64 * 65];
  const float s = -1.0f / reg[0];
  const int bi = blockIdx.x & 63;   // row tile
  const int bj = blockIdx.x >> 6;   // col tile
  const int tid = threadIdx.x;

  for (int idx = tid; idx < 64 * 16; idx += 256) {
    const int r  = idx >> 4;
    const int c4 = (idx & 15) << 2;
    const size_t g = (size_t)(bi * 64 + r) * NN + (size_t)bj * 64 + c4;
    float4 m4 = *(const float4*)(M + g);
    float4 e;
    e.x = expf(m4.x * s);
    e.y = expf(m4.y * s);
    e.z = expf(m4.z * s);
    e.w = expf(m4.w * s);
    *(float4*)(C + g) = e;
    tile[r * 65 + c4 + 0] = e.x;
    tile[r * 65 + c4 + 1] = e.y;
    tile[r * 65 + c4 + 2] = e.z;
    tile[r * 65 + c4 + 3] = e.w;
  }
  __syncthreads();
  for (int idx = tid; idx < 64 * 16; idx += 256) {
    const int r  = idx >> 4;
    const int c4 = (idx & 15) << 2;
    float4 o;
    o.x = tile[(c4 + 0) * 65 + r];
    o.y = tile[(c4 + 1) * 65 + r];
    o.z = tile[(c4 + 2) * 65 + r];
    o.w = tile[(c4 + 3) * 65 + r];
    *(float4*)(CT + (size_t)(bj * 64 + r) * NN + (size_t)bi * 64 + c4) = o;
  }
}

// ---------------------------------------------------------------------------
// Partial GEMM: P[ks] (4096x64) = A[:, ks-range] @ Bp[ks-range, :] (packed B).
// Workgroup: 8 waves, 128x64 output block; wave w owns rows [w*16, w*16+16).
// Double-buffered async LDS staging:
//   As stride 68 dwords  -> A-operand ds_load_b64 hits all 64 banks (4m+t)
//   Bs pair-row stride 160 dwords -> half-wave k / k+2 groups on disjoint
//     32-bank halves (160 % 64 == 32); each B operand = one ds_load_b64.
// 12 async b128 ops per thread per chunk; wait asynccnt<=12 keeps the next
// chunk's copies in flight while the current chunk feeds 64 WMMAs.
// ---------------------------------------------------------------------------
#define ASTR  68
#define BPSTR 160

__global__ __launch_bounds__(256) void gemm_n64_part(
    const float* __restrict__ A,     // NN x NN row-major
    const float* __restrict__ Bp,    // NN x NBB, k-pair packed
    float* __restrict__ P)           // KSPLIT x NN x NBB partials (linear)
{
  __shared__ float As[2][128 * ASTR];
  __shared__ float Bs[2][(CHUNK / 2) * BPSTR];

  const int tid  = threadIdx.x;
  const int wave = tid >> 5;
  const int lane = tid & 31;
  const int m    = lane & 15;          // row in 16 (A) / col in 16 (B)
  const int kb   = (lane >> 4) << 1;   // 0 or 2 : K sub-block per half-wave
  const int rowBase = blockIdx.x * 128;
  const int kBase   = blockIdx.y * (NN / KSPLIT);

  // async-issue one 64-wide K chunk into LDS buffer `buf`
  auto issue = [&](int chunk, int buf) {
    const int k0 = kBase + chunk * CHUNK;
#pragma unroll
    for (int q = 0; q < 8; ++q) {            // A: 128 x 64 = 2048 granules
      const int idx = tid + q * 256;
      const int r  = idx >> 4;
      const int c4 = (idx & 15) << 2;
      cp16(A + (size_t)(rowBase + r) * NN + k0 + c4, &As[buf][r * ASTR + c4]);
    }
#pragma unroll
    for (int q = 0; q < 4; ++q) {            // B: 32 pair-rows x 32 granules
      const int idx = tid + q * 256;
      const int pr = idx >> 5;               // pair-row 0..31
      const int c4 = (idx & 31) << 2;        // float offset 0..124
      cp16(Bp + (size_t)((k0 >> 1) + pr) * 128 + c4,
           &Bs[buf][pr * BPSTR + c4]);
    }
  };

  v8f acc0 = {}, acc1 = {}, acc2 = {}, acc3 = {};

  issue(0, 0);
  for (int i = 0; i < CHUNKS; ++i) {
    const int cur = i & 1;
    if (i + 1 < CHUNKS) {
      issue(i + 1, cur ^ 1);
      wait_async_le<12>();   // current chunk done; next 12 still in flight
    } else {
      wait_async_le<0>();
    }
    __syncthreads();

    const float* Asc = As[cur];
    const float* Bsc = Bs[cur];
#pragma unroll
    for (int k2 = 0; k2 < CHUNK; k2 += 4) {
      const int k = k2 + kb;                                   // even
      v2f a = *(const v2f*)(Asc + (wave * 16 + m) * ASTR + k); // A[m][k], [k+1]
      const float* bp = Bsc + (k >> 1) * BPSTR + 2 * m;        // {B[k][n],B[k+1][n]}
      v2f b0 = *(const v2f*)(bp + 0);                          // n = m
      v2f b1 = *(const v2f*)(bp + 32);                         // n = m + 16
      v2f b2 = *(const v2f*)(bp + 64);                         // n = m + 32
      v2f b3 = *(const v2f*)(bp + 96);                         // n = m + 48
      acc0 = __builtin_amdgcn_wmma_f32_16x16x4_f32(false, a, false, b0,
                                                   (short)0, acc0, false, false);
      acc1 = __builtin_amdgcn_wmma_f32_16x16x4_f32(false, a, false, b1,
                                                   (short)0, acc1, false, false);
      acc2 = __builtin_amdgcn_wmma_f32_16x16x4_f32(false, a, false, b2,
                                                   (short)0, acc2, false, false);
      acc3 = __builtin_amdgcn_wmma_f32_16x16x4_f32(false, a, false, b3,
                                                   (short)0, acc3, false, false);
    }
    __syncthreads();   // all waves done reading buf[cur] before it is refilled
  }

  // Store partials (linear). C/D layout: VGPR i -> row i / i+8 per half-wave.
  float* out = P + (size_t)blockIdx.y * NN * NBB;
  const int rOff = (lane >> 4) << 3;   // 0 or 8
#pragma unroll
  for (int i = 0; i < 8; i++) {
    const int r = rowBase + wave * 16 + rOff + i;
    const size_t base = (size_t)r * NBB + m;
    out[base + 0]  = acc0[i];
    out[base + 16] = acc1[i];
    out[base + 32] = acc2[i];
    out[base + 48] = acc3[i];
  }
}

// ---------------------------------------------------------------------------
// Deterministic fixed-order K-split reduction.
//   DIVNUM=1: D = num / sum, written k-pair packed (feeds next GEMM's B).
//   DIVNUM=0: D = sum, written linear (feeds rowpow).
// ---------------------------------------------------------------------------
template <int DIVNUM>
__global__ __launch_bounds__(256) void reduce_ep(
    const float* __restrict__ P, const float* __restrict__ num,
    float* __restrict__ D)
{
  const int i = blockIdx.x * 256 + threadIdx.x;
  float s = 0.0f;
#pragma unroll
  for (int k = 0; k < KSPLIT; ++k) s += P[(size_t)k * NN * NBB + i];
  if constexpr (DIVNUM) {
    const int row = i >> 6, col = i & 63;
    D[packIdx(row, col)] = num[i] / s;
  } else {
    D[i] = s;
  }
}

// ---------------------------------------------------------------------------
// Per-row: yhat = exp(sum_j lbd[j]*log v[row][j]) ; u = yhat / v.
// v read linear; u written k-pair packed (feeds GEMM1's B operand).
// ---------------------------------------------------------------------------
__global__ __launch_bounds__(64) void rowpow(
    const float* __restrict__ v, const float* __restrict__ lbd,
    float* __restrict__ yhat, float* __restrict__ u)
{
  __shared__ float red[64];
  const int row = blockIdx.x;
  const int j   = threadIdx.x;
  const float val = v[(size_t)row * NBB + j];
  red[j] = lbd[j] * logf(val);
  __syncthreads();
  for (int s = 32; s > 0; s >>= 1) {
    if (j < s) red[j] += red[j + s];
    __syncthreads();
  }
  const float yh = expf(red[0]);
  u[packIdx(row, j)] = yh / val;
  if (j == 0) yhat[row] = yh;
}

__global__ __launch_bounds__(256) void fill_ones(float* __restrict__ p, int n) {
  int i = blockIdx.x * 256 + threadIdx.x;
  if (i < n) p[i] = 1.0f;
}

__global__ __launch_bounds__(1024) void finalize(
    const float* __restrict__ yhat, const float* __restrict__ a,
    float* __restrict__ out)
{
  __shared__ float red[1024];
  const int t = threadIdx.x;
  float s = 0.0f;
  for (int i = t; i < NN; i += 1024) {
    const float d = yhat[i] - a[i];
    s += d * d;
  }
  red[t] = s;
  __syncthreads();
  for (int k = 512; k > 0; k >>= 1) {
    if (t < k) red[t] += red[t + k];
    __syncthreads();
  }
  if (t == 0) out[0] = red[0];
}

// ---------------------------------------------------------------------------
extern "C" void kernel_launch(void* const* d_in, const int* in_sizes, int n_in,
                              void* d_out, int out_size, void* d_ws, size_t ws_size,
                              hipStream_t stream)
{
  const float* a   = (const float*)d_in[0];   // (N,1)
  const float* M   = (const float*)d_in[1];   // (N,N)
  const float* b   = (const float*)d_in[2];   // (N,NBB)
  const float* lbd = (const float*)d_in[3];   // (NBB,1)
  const float* reg = (const float*)d_in[4];   // scalar

  float* C    = (float*)d_ws;                 // 64 MB
  float* CT   = C  + (size_t)NN * NN;         // 64 MB
  float* u    = CT + (size_t)NN * NN;         // 1 MB  (k-pair packed)
  float* t    = u  + (size_t)NN * NBB;        // 1 MB  (k-pair packed)
  float* vcur = t  + (size_t)NN * NBB;        // 1 MB  (linear)
  float* yh   = vcur + (size_t)NN * NBB;      // 16 KB
  float* P    = yh + NN;                      // 4 MB partials (linear)

  build_c_ct<<<64 * 64, 256, 0, stream>>>(M, reg, C, CT);
  fill_ones<<<(NN * NBB + 255) / 256, 256, 0, stream>>>(u, NN * NBB);

  const dim3 ggrid(NN / 128, KSPLIT);
  for (int it = 0; it < ITERS; ++it) {
    // t = b / (C @ u)
    gemm_n64_part<<<ggrid, 256, 0, stream>>>(C, u, P);
    reduce_ep<1><<<NN * NBB / 256, 256, 0, stream>>>(P, b, t);
    // v = C^T @ t
    gemm_n64_part<<<ggrid, 256, 0, stream>>>(CT, t, P);
    reduce_ep<0><<<NN * NBB / 256, 256, 0, stream>>>(P, nullptr, vcur);
    // yhat = prod_j v^lbd ; u = yhat / v
    rowpow<<<NN, 64, 0, stream>>>(vcur, lbd, yh, u);
  }

  finalize<<<1, 1024, 0, stream>>>(yh, a, (float*)d_out);
}